// Decoder_base_87393994539039
// MI455X (gfx1250) — compile-verified
//
#include <hip/hip_runtime.h>
#include <math.h>

typedef __attribute__((ext_vector_type(2))) float v2f;
typedef __attribute__((ext_vector_type(8))) float v8f;

namespace {
constexpr int Bsz = 8;
constexpr int Fdim = 32;
constexpr int N3v = 128, N2v = 1024, N1v = 8192, N0v = 65536;
constexpr int E3v = 768, E2v = 6144, E1v = 49152, E0v = 393216;
constexpr int Tv = 131072;
}

// ---------------- dense GEMM head: out[8,N] = A[8,K] @ W[K,N] + bias --------
// One wave per 32-column tile (two 16x16 WMMA accumulators), fp32 WMMA
// 16x16x4. M padded 8->16: lanes m>=8 read row m&7 (valid memory); the
// duplicate output rows are discarded at store time, so no guard is needed.
__global__ void gemm8_wmma(const float* __restrict__ A, const float* __restrict__ W,
                           const float* __restrict__ bias, float* __restrict__ out,
                           int K, int N) {
  int wave = (int)((blockIdx.x * blockDim.x + threadIdx.x) >> 5);
  int lane = (int)(threadIdx.x & 31);
  int ntiles = N >> 5;
  if (wave >= ntiles) return;
  int m = lane & 15;   // A row / D column index
  int g = lane >> 4;   // K-half selector / D row-half selector
  int col0 = (wave << 5) + m;
  int col1 = col0 + 16;
  float bc0 = bias[col0], bc1 = bias[col1];
  v8f acc0 = {bc0, bc0, bc0, bc0, bc0, bc0, bc0, bc0};
  v8f acc1 = {bc1, bc1, bc1, bc1, bc1, bc1, bc1, bc1};
  const float* arow = A + (size_t)(m & 7) * K + 2 * g;  // no branch: rows 8..15 duplicate 0..7
  const float* w0 = W + (size_t)(2 * g) * N;
#pragma unroll 4
  for (int k4 = 0; k4 < K; k4 += 4) {
    v2f a = *reinterpret_cast<const v2f*>(arow + k4);
    v2f bA = { w0[col0], w0[(size_t)N + col0] };
    v2f bB = { w0[col1], w0[(size_t)N + col1] };
    w0 += (size_t)4 * N;
    acc0 = __builtin_amdgcn_wmma_f32_16x16x4_f32(false, a, false, bA, (short)0,
                                                 acc0, false, false);
    acc1 = __builtin_amdgcn_wmma_f32_16x16x4_f32(false, a, false, bB, (short)0,
                                                 acc1, false, false);
  }
  if (g == 0) {
#pragma unroll
    for (int r = 0; r < 8; ++r) {
      out[(size_t)r * N + col0] = acc0[r];
      out[(size_t)r * N + col1] = acc1[r];
    }
  }
}

// --------------- BatchNorm1d (training stats over batch of 8) + ReLU --------
__global__ void bn_relu8(float* __restrict__ h, const float* __restrict__ gain,
                         const float* __restrict__ beta, int N) {
  int j = (int)(blockIdx.x * blockDim.x + threadIdx.x);
  if (j >= N) return;
  float v[8];
  float mu = 0.0f;
#pragma unroll
  for (int b = 0; b < 8; ++b) { v[b] = h[(size_t)b * N + j]; mu += v[b]; }
  mu *= 0.125f;
  float var = 0.0f;
#pragma unroll
  for (int b = 0; b < 8; ++b) { float d = v[b] - mu; var += d * d; }
  var *= 0.125f;
  float sc = gain[j] / sqrtf(var + 1e-5f);
  float bo = beta[j];
#pragma unroll
  for (int b = 0; b < 8; ++b) {
    float o = (v[b] - mu) * sc + bo;
    h[(size_t)b * N + j] = o > 0.0f ? o : 0.0f;
  }
}

// ---------------- graph degree + symmetric-norm edge weights ----------------
__global__ void deg_count(const int* __restrict__ dst, float* deg, int E) {
  int e = (int)(blockIdx.x * blockDim.x + threadIdx.x);
  if (e < E) atomicAdd(&deg[dst[e]], 1.0f);
}

__global__ void edge_norm(const int* __restrict__ src, const int* __restrict__ dst,
                          const float* __restrict__ deg, float* __restrict__ nrm, int E) {
  int e = (int)(blockIdx.x * blockDim.x + threadIdx.x);
  if (e >= E) return;
  float ds = deg[src[e]], dd = deg[dst[e]];
  float is = ds > 0.0f ? 1.0f / sqrtf(ds) : 0.0f;
  float id = dd > 0.0f ? 1.0f / sqrtf(dd) : 0.0f;
  nrm[e] = -(is * id);
}

// ---------------- edge scatter: t[:,dst,:] += x[:,src,:] * norm[e] ----------
// One lane per (edge, feature); a wave covers all 32 features of one edge.
__global__ void scatter_prop(const float* __restrict__ x, float* t,
                             const int* __restrict__ src, const int* __restrict__ dst,
                             const float* __restrict__ nrm, int E, int n) {
  unsigned id = blockIdx.x * blockDim.x + threadIdx.x;
  unsigned e = id >> 5;
  unsigned f = id & 31u;
  if (e >= (unsigned)E) return;
  int s = src[e], d = dst[e];
  float w = nrm[e];
  size_t so = (size_t)s * Fdim + f;
  size_t dof = (size_t)d * Fdim + f;
  size_t stride = (size_t)n * Fdim;
#pragma unroll
  for (int b = 0; b < Bsz; ++b)
    atomicAdd(&t[dof + (size_t)b * stride], x[so + (size_t)b * stride] * w);
}

// ---------------- t2 = 2*prop(t1) - x (in-place on the prop buffer) ---------
__global__ void combine_t2(float* __restrict__ t2, const float* __restrict__ x, int count) {
  int i = (int)(blockIdx.x * blockDim.x + threadIdx.x);
  if (i < count) t2[i] = 2.0f * t2[i] - x[i];
}

// ---------------- Cheb feature transform via fp32 WMMA ----------------------
// out[r, 0:32] = bias + [x|t1|t2][r, 0:96] @ Wcat[96,32]; optional ReLU.
// out may alias t2 (row-local computation). One wave per 16-row tile,
// two 16x16 accumulators covering the 32 output columns, K-loop of 24 WMMAs.
__global__ void cheb_transform_wmma(const float* __restrict__ x,
                                    const float* __restrict__ t1,
                                    const float* t2, float* out,
                                    const float* __restrict__ W,     // [96][32]
                                    const float* __restrict__ bias,  // [32]
                                    int rows, int relu) {
  __shared__ float wl[96 * 32];
  for (int i = (int)threadIdx.x; i < 96 * 32; i += (int)blockDim.x) wl[i] = W[i];
  __syncthreads();

  int wave = (int)((blockIdx.x * blockDim.x + threadIdx.x) >> 5);
  int lane = (int)(threadIdx.x & 31);
  int rowBase = wave << 4;
  if (rowBase >= rows) return;
  int m = lane & 15;
  int g = lane >> 4;

  float b0 = bias[m], b1 = bias[m + 16];
  v8f acc0 = {b0, b0, b0, b0, b0, b0, b0, b0};
  v8f acc1 = {b1, b1, b1, b1, b1, b1, b1, b1};

  const float* sp[3] = {x, t1, t2};
#pragma unroll
  for (int seg = 0; seg < 3; ++seg) {
    const float* arow = sp[seg] + (size_t)(rowBase + m) * Fdim + 2 * g;
#pragma unroll
    for (int q = 0; q < 8; ++q) {
      v2f a = *reinterpret_cast<const v2f*>(arow + q * 4);
      int wr = (seg * 8 + q) * 4 + 2 * g;
      v2f bA = { wl[wr * 32 + m],      wl[wr * 32 + 32 + m] };
      v2f bB = { wl[wr * 32 + m + 16], wl[wr * 32 + 48 + m] };
      acc0 = __builtin_amdgcn_wmma_f32_16x16x4_f32(false, a, false, bA, (short)0,
                                                   acc0, false, false);
      acc1 = __builtin_amdgcn_wmma_f32_16x16x4_f32(false, a, false, bB, (short)0,
                                                   acc1, false, false);
    }
  }
#pragma unroll
  for (int r = 0; r < 8; ++r) {
    int row = rowBase + r + 8 * g;
    float u0 = acc0[r], u1 = acc1[r];
    if (relu) { u0 = fmaxf(u0, 0.0f); u1 = fmaxf(u1, 0.0f); }
    out[(size_t)row * Fdim + m] = u0;
    out[(size_t)row * Fdim + m + 16] = u1;
  }
}

// ---------------- barycentric upsample: fine[b,n,f] = sum_k xc[b,idx[n,k],f]*w[n,k]
__global__ void upsample_k(const float* __restrict__ xc, float* __restrict__ xf,
                           const int* __restrict__ idx, const float* __restrict__ w,
                           int Nf, int Nc) {
  size_t id = (size_t)blockIdx.x * blockDim.x + threadIdx.x;
  size_t tot = (size_t)Bsz * Nf * Fdim;
  if (id >= tot) return;
  unsigned f = (unsigned)(id & 31u);
  size_t rn = id >> 5;
  int n = (int)(rn % (size_t)Nf);
  int b = (int)(rn / (size_t)Nf);
  const int* ix = idx + (size_t)n * 3;
  const float* ww = w + (size_t)n * 3;
  size_t base = (size_t)b * Nc * Fdim + f;
  float acc = xc[base + (size_t)ix[0] * Fdim] * ww[0] +
              xc[base + (size_t)ix[1] * Fdim] * ww[1] +
              xc[base + (size_t)ix[2] * Fdim] * ww[2];
  xf[id] = acc;
}

// ---------------- final Cheb transform with O=3 outputs ---------------------
__global__ void cheb_out3(const float* __restrict__ x, const float* __restrict__ t1,
                          const float* __restrict__ t2, const float* __restrict__ W,
                          const float* __restrict__ bias, float* __restrict__ out,
                          int rows) {
  __shared__ float wl[96 * 3];
  __shared__ float bl[3];
  for (int i = (int)threadIdx.x; i < 96 * 3; i += (int)blockDim.x) wl[i] = W[i];
  if (threadIdx.x < 3) bl[threadIdx.x] = bias[threadIdx.x];
  __syncthreads();
  int r = (int)(blockIdx.x * blockDim.x + threadIdx.x);
  if (r >= rows) return;
  float a0 = bl[0], a1 = bl[1], a2 = bl[2];
  const float* xr = x + (size_t)r * Fdim;
  const float* t1r = t1 + (size_t)r * Fdim;
  const float* t2r = t2 + (size_t)r * Fdim;
#pragma unroll
  for (int f = 0; f < 32; ++f) {
    float v = xr[f];
    a0 += v * wl[f * 3 + 0]; a1 += v * wl[f * 3 + 1]; a2 += v * wl[f * 3 + 2];
  }
#pragma unroll
  for (int f = 0; f < 32; ++f) {
    float v = t1r[f];
    a0 += v * wl[(32 + f) * 3 + 0]; a1 += v * wl[(32 + f) * 3 + 1]; a2 += v * wl[(32 + f) * 3 + 2];
  }
#pragma unroll
  for (int f = 0; f < 32; ++f) {
    float v = t2r[f];
    a0 += v * wl[(64 + f) * 3 + 0]; a1 += v * wl[(64 + f) * 3 + 1]; a2 += v * wl[(64 + f) * 3 + 2];
  }
  out[(size_t)r * 3 + 0] = a0;
  out[(size_t)r * 3 + 1] = a1;
  out[(size_t)r * 3 + 2] = a2;
}

// ---------------- signed volume per batch (deterministic reduction) ---------
__global__ void tri_volume(const float* __restrict__ x3, const int* __restrict__ tri,
                           float* __restrict__ volsum) {
  __shared__ float red[512];
  int b = (int)blockIdx.x;
  const float* xb = x3 + (size_t)b * N0v * 3;
  float acc = 0.0f;
  for (int t = (int)threadIdx.x; t < Tv; t += 512) {
    int i0 = tri[t * 3 + 0], i1 = tri[t * 3 + 1], i2 = tri[t * 3 + 2];
    const float* p0 = xb + (size_t)i0 * 3;
    const float* p1 = xb + (size_t)i1 * 3;
    const float* p2 = xb + (size_t)i2 * 3;
    float cx = p1[1] * p2[2] - p1[2] * p2[1];
    float cy = p1[2] * p2[0] - p1[0] * p2[2];
    float cz = p1[0] * p2[1] - p1[1] * p2[0];
    acc += p0[0] * cx + p0[1] * cy + p0[2] * cz;
  }
  red[threadIdx.x] = acc;
  __syncthreads();
  for (int s = 256; s > 0; s >>= 1) {
    if ((int)threadIdx.x < s) red[threadIdx.x] += red[threadIdx.x + s];
    __syncthreads();
  }
  if (threadIdx.x == 0) volsum[b] = red[0] / 6.0f;
}

// ---------------- centroid: c[b,f] = mean over nodes ------------------------
__global__ void mean_c(const float* __restrict__ x3, float* __restrict__ cmean) {
  __shared__ float red[256];
  int b = (int)blockIdx.x / 3;
  int f = (int)blockIdx.x % 3;
  float acc = 0.0f;
  for (int n = (int)threadIdx.x; n < N0v; n += 256)
    acc += x3[((size_t)b * N0v + n) * 3 + f];
  red[threadIdx.x] = acc;
  __syncthreads();
  for (int s = 128; s > 0; s >>= 1) {
    if ((int)threadIdx.x < s) red[threadIdx.x] += red[threadIdx.x + s];
    __syncthreads();
  }
  if (threadIdx.x == 0) cmean[blockIdx.x] = red[0] * (1.0f / (float)N0v);
}

// ---------------- out = barycenter + (x - c) * cbrt(vol0/(vol+eps)) ---------
__global__ void finalize_k(const float* __restrict__ x3, const float* __restrict__ cmean,
                           const float* __restrict__ vol, const float* __restrict__ bary,
                           const float* __restrict__ volin, float* __restrict__ out) {
  size_t id = (size_t)blockIdx.x * blockDim.x + threadIdx.x;
  size_t tot = (size_t)Bsz * N0v * 3;
  if (id >= tot) return;
  int f = (int)(id % 3);
  int b = (int)(id / ((size_t)N0v * 3));
  float s = cbrtf(volin[0] / (vol[b] + 1e-6f));
  out[id] = bary[f] + (x3[id] - cmean[b * 3 + f]) * s;
}

// ============================================================================
extern "C" void kernel_launch(void* const* d_in, const int* in_sizes, int n_in,
                              void* d_out, int out_size, void* d_ws, size_t ws_size,
                              hipStream_t stream) {
  (void)in_sizes; (void)n_in; (void)out_size; (void)ws_size;
  const float* z       = (const float*)d_in[0];
  const int*   adj0    = (const int*)d_in[1];
  const int*   adj1    = (const int*)d_in[2];
  const int*   adj2    = (const int*)d_in[3];
  const int*   adj3    = (const int*)d_in[4];
  const int*   up_idx0 = (const int*)d_in[5];
  const int*   up_idx1 = (const int*)d_in[6];
  const int*   up_idx2 = (const int*)d_in[7];
  const float* up_w0   = (const float*)d_in[8];
  const float* up_w1   = (const float*)d_in[9];
  const float* up_w2   = (const float*)d_in[10];
  const int*   tris    = (const int*)d_in[11];
  const float* bary    = (const float*)d_in[12];
  const float* volin   = (const float*)d_in[13];
  // params flattened depth-first in dict insertion order
  const float* W1   = (const float*)d_in[14];
  const float* b1   = (const float*)d_in[15];
  const float* l2W  = (const float*)d_in[16];
  const float* l2b  = (const float*)d_in[17];
  const float* l2g  = (const float*)d_in[18];
  const float* l2be = (const float*)d_in[19];
  const float* l3W  = (const float*)d_in[20];
  const float* l3b  = (const float*)d_in[21];
  const float* l3g  = (const float*)d_in[22];
  const float* l3be = (const float*)d_in[23];
  const float* l4W  = (const float*)d_in[24];
  const float* l4b  = (const float*)d_in[25];
  const float* l4g  = (const float*)d_in[26];
  const float* l4be = (const float*)d_in[27];
  const float* c5W  = (const float*)d_in[28];
  const float* c5b  = (const float*)d_in[29];
  const float* c6Wc = (const float*)d_in[30];
  const float* c6bc = (const float*)d_in[31];
  const float* c6Wf = (const float*)d_in[32];
  const float* c6bf = (const float*)d_in[33];
  const float* c7Wc = (const float*)d_in[34];
  const float* c7bc = (const float*)d_in[35];
  const float* c7Wf = (const float*)d_in[36];
  const float* c7bf = (const float*)d_in[37];
  const float* c8Wc = (const float*)d_in[38];
  const float* c8bc = (const float*)d_in[39];
  const float* c8Wf = (const float*)d_in[40];
  const float* c8bf = (const float*)d_in[41];
  const float* c9W  = (const float*)d_in[42];
  const float* c9b  = (const float*)d_in[43];

  // ---------------- workspace layout ----------------
  char* wsb = (char*)d_ws;
  const size_t BIG = (size_t)Bsz * N0v * Fdim * sizeof(float);  // 64 MB
  float* bb0 = (float*)(wsb);
  float* bb1 = (float*)(wsb + BIG);
  float* bb2 = (float*)(wsb + 2 * BIG);
  char* p = wsb + 3 * BIG;
  auto alloc = [&](size_t bytes) -> float* {
    float* r = (float*)p;
    p += (bytes + 255) & ~(size_t)255;
    return r;
  };
  float* deg  = alloc((size_t)N0v * 4);
  float* nrm0 = alloc((size_t)E0v * 4);
  float* nrm1 = alloc((size_t)E1v * 4);
  float* nrm2 = alloc((size_t)E2v * 4);
  float* nrm3 = alloc((size_t)E3v * 4);
  float* hA   = alloc((size_t)Bsz * 4096 * 4);
  float* hB   = alloc((size_t)Bsz * 4096 * 4);
  float* x3   = alloc((size_t)Bsz * N0v * 3 * 4);
  float* volsum = alloc(8 * 4);
  float* cmean  = alloc(24 * 4);

  struct Lv { const int* src; const int* dst; const float* nrm; int n; int E; };
  Lv lv[4] = {
    {adj0, adj0 + E0v, nrm0, N0v, E0v},
    {adj1, adj1 + E1v, nrm1, N1v, E1v},
    {adj2, adj2 + E2v, nrm2, N2v, E2v},
    {adj3, adj3 + E3v, nrm3, N3v, E3v},
  };

  auto cdiv = [](size_t a, size_t b) -> unsigned { return (unsigned)((a + b - 1) / b); };

  // ---------------- per-level symmetric-norm edge weights ----------------
  for (int L = 0; L < 4; ++L) {
    hipMemsetAsync(deg, 0, (size_t)lv[L].n * 4, stream);
    deg_count<<<cdiv(lv[L].E, 256), 256, 0, stream>>>(lv[L].dst, deg, lv[L].E);
    edge_norm<<<cdiv(lv[L].E, 256), 256, 0, stream>>>(lv[L].src, lv[L].dst, deg,
                                                      (float*)lv[L].nrm, lv[L].E);
  }

  // ---------------- dense head: W1 then 3x (Linear + BN + ReLU) ----------
  gemm8_wmma<<<32, 128, 0, stream>>>(z, W1, b1, hA, 128, 4096);
  gemm8_wmma<<<32, 128, 0, stream>>>(hA, l2W, l2b, hB, 4096, 4096);
  bn_relu8<<<16, 256, 0, stream>>>(hB, l2g, l2be, 4096);
  gemm8_wmma<<<32, 128, 0, stream>>>(hB, l3W, l3b, hA, 4096, 4096);
  bn_relu8<<<16, 256, 0, stream>>>(hA, l3g, l3be, 4096);
  gemm8_wmma<<<32, 128, 0, stream>>>(hA, l4W, l4b, hB, 4096, 4096);
  bn_relu8<<<16, 256, 0, stream>>>(hB, l4g, l4be, 4096);
  // hB is now the level-3 activation tensor [8,128,32] (same linear layout)

  // ---------------- ChebConv helper ----------------
  auto run_cheb = [&](const float* x, float* s1, float* s2, int L,
                      const float* W, const float* bias, int relu) {
    const Lv& lvv = lv[L];
    size_t cnt = (size_t)Bsz * lvv.n * Fdim;
    unsigned sb = cdiv((size_t)lvv.E * 32, 256);
    hipMemsetAsync(s1, 0, cnt * 4, stream);
    scatter_prop<<<sb, 256, 0, stream>>>(x, s1, lvv.src, lvv.dst, lvv.nrm, lvv.E, lvv.n);
    hipMemsetAsync(s2, 0, cnt * 4, stream);
    scatter_prop<<<sb, 256, 0, stream>>>(s1, s2, lvv.src, lvv.dst, lvv.nrm, lvv.E, lvv.n);
    combine_t2<<<cdiv(cnt, 256), 256, 0, stream>>>(s2, x, (int)cnt);
    int rows = Bsz * lvv.n;
    cheb_transform_wmma<<<(unsigned)(rows / 64), 128, 0, stream>>>(
        x, s1, s2, s2, W, bias, rows, relu);  // output written into s2
  };
  auto run_up = [&](const float* xc, float* xf, const int* idx, const float* w,
                    int Nf, int Nc) {
    size_t tot = (size_t)Bsz * Nf * Fdim;
    upsample_k<<<cdiv(tot, 256), 256, 0, stream>>>(xc, xf, idx, w, Nf, Nc);
  };

  // cheb5 (level 3, no relu): x = hB, result -> bb1
  run_cheb(hB, bb0, bb1, 3, c5W, c5b, 0);
  // ctu6: coarse cheb on level 3 (relu) -> bb2; upsample N3->N2 -> bb0; fine cheb level 2 -> bb2
  run_cheb(bb1, bb0, bb2, 3, c6Wc, c6bc, 1);
  run_up(bb2, bb0, up_idx2, up_w2, N2v, N3v);
  run_cheb(bb0, bb1, bb2, 2, c6Wf, c6bf, 1);
  // ctu7: coarse level 2 -> bb1; upsample N2->N1 -> bb0; fine level 1 -> bb2
  run_cheb(bb2, bb0, bb1, 2, c7Wc, c7bc, 1);
  run_up(bb1, bb0, up_idx1, up_w1, N1v, N2v);
  run_cheb(bb0, bb1, bb2, 1, c7Wf, c7bf, 1);
  // ctu8: coarse level 1 -> bb1; upsample N1->N0 -> bb0; fine level 0 -> bb2
  run_cheb(bb2, bb0, bb1, 1, c8Wc, c8bc, 1);
  run_up(bb1, bb0, up_idx0, up_w0, N0v, N1v);
  run_cheb(bb0, bb1, bb2, 0, c8Wf, c8bf, 1);

  // cheb9 (level 0, O=3, no relu): props on bb2 into bb0/bb1, transform -> x3
  {
    const Lv& lvv = lv[0];
    size_t cnt = (size_t)Bsz * lvv.n * Fdim;
    unsigned sb = cdiv((size_t)lvv.E * 32, 256);
    hipMemsetAsync(bb0, 0, cnt * 4, stream);
    scatter_prop<<<sb, 256, 0, stream>>>(bb2, bb0, lvv.src, lvv.dst, lvv.nrm, lvv.E, lvv.n);
    hipMemsetAsync(bb1, 0, cnt * 4, stream);
    scatter_prop<<<sb, 256, 0, stream>>>(bb0, bb1, lvv.src, lvv.dst, lvv.nrm, lvv.E, lvv.n);
    combine_t2<<<cdiv(cnt, 256), 256, 0, stream>>>(bb1, bb2, (int)cnt);
    int rows = Bsz * lvv.n;
    cheb_out3<<<cdiv(rows, 256), 256, 0, stream>>>(bb2, bb0, bb1, c9W, c9b, x3, rows);
  }

  // ---------------- volume, centroid, finalize ----------------
  tri_volume<<<8, 512, 0, stream>>>(x3, tris, volsum);
  mean_c<<<24, 256, 0, stream>>>(x3, cmean);
  size_t tot = (size_t)Bsz * N0v * 3;
  finalize_k<<<cdiv(tot, 256), 256, 0, stream>>>(x3, cmean, volsum, bary, volin,
                                                 (float*)d_out);
}